// DigitCaps_27822798143999
// MI455X (gfx1250) — compile-verified
//
#include <hip/hip_runtime.h>
#include <math.h>

#define ROUTES 1152
#define CAPS 10
#define IN_CH 8
#define OUT_CH 16
#define BATCH 256
#define NUM_ITER 3
#define KDIM (ROUTES * IN_CH)   // 9216
#define NDIM (CAPS * OUT_CH)    // 160
#define RC (ROUTES * CAPS)      // 11520
#define KBLK (KDIM / 4)         // 2304 K-blocks of 4
#define KSPLIT 2                // K split across blockIdx.z
#define NCHUNK (KSPLIT * 8)     // 16 wave-chunks over K
#define KB_PER_WAVE (KBLK / NCHUNK)  // 144 WMMAs per wave

typedef __attribute__((ext_vector_type(2))) float v2f;
typedef __attribute__((ext_vector_type(8))) float v8f;

// ---------------------------------------------------------------------------
// zero-init b_ij (harness poisons ws; no cross-call state allowed)
// ---------------------------------------------------------------------------
__global__ void zero_kernel(float* __restrict__ p, int n) {
    int i = blockIdx.x * blockDim.x + threadIdx.x;
    if (i < n) p[i] = 0.0f;
}

// ---------------------------------------------------------------------------
// softmax over routes (axis 0) of b_ij[R,C] -> c_ij[R,C]; one block per cap
// ---------------------------------------------------------------------------
__global__ void __launch_bounds__(256) softmax_kernel(const float* __restrict__ b,
                                                      float* __restrict__ c) {
    const int cap = blockIdx.x;
    const int tid = threadIdx.x;
    __shared__ float sd[256];
    float mx = -3.4e38f;
    for (int r = tid; r < ROUTES; r += 256) mx = fmaxf(mx, b[r * CAPS + cap]);
    sd[tid] = mx; __syncthreads();
    for (int o = 128; o > 0; o >>= 1) {
        if (tid < o) sd[tid] = fmaxf(sd[tid], sd[tid + o]);
        __syncthreads();
    }
    mx = sd[0]; __syncthreads();
    float sum = 0.0f;
    for (int r = tid; r < ROUTES; r += 256) sum += expf(b[r * CAPS + cap] - mx);
    sd[tid] = sum; __syncthreads();
    for (int o = 128; o > 0; o >>= 1) {
        if (tid < o) sd[tid] += sd[tid + o];
        __syncthreads();
    }
    const float inv = 1.0f / sd[0];
    for (int r = tid; r < ROUTES; r += 256)
        c[r * CAPS + cap] = expf(b[r * CAPS + cap] - mx) * inv;
}

// ---------------------------------------------------------------------------
// Build Wc directly in WMMA B-fragment order:
//   wcs[((ntile*KBLK + kb)*32 + lane)*2 + j] = B[k, n]
// with k = kb*4 + (lane>>4)*2 + j, n = ntile*16 + (lane&15),
//      B[k,n] = c_ij[r, ntile] * W[r, ntile, o=n&15, i=k&7],  r = k>>3.
// gemm_s then reads one coalesced b64 per lane per WMMA step.
// ---------------------------------------------------------------------------
__global__ void scale_kernel(const float* __restrict__ W, const float* __restrict__ c,
                             float* __restrict__ wcs) {
    int t = blockIdx.x * blockDim.x + threadIdx.x;
    if (t >= KDIM * NDIM) return;
    const int j     = t & 1;
    const int lane  = (t >> 1) & 31;
    const int kb    = (t >> 6) % KBLK;
    const int ntile = (t >> 6) / KBLK;          // 0..9 == caps index
    const int k = kb * 4 + (lane >> 4) * 2 + j; // global K index
    const int r = k >> 3;
    const int i = k & 7;
    const int o = lane & 15;
    wcs[t] = c[r * CAPS + ntile] *
             W[(((size_t)(r * CAPS + ntile)) * OUT_CH + o) * IN_CH + i];
}

// ---------------------------------------------------------------------------
// s_part[z][256 x 160] = partial of x_flat[256 x 9216] @ Wc[9216 x 160]
// grid (10, 16, KSPLIT), 8 waves/block; inner loop = 2x b64 load + 1x wmma.
// Deterministic: LDS tree reduce in-block, fixed-order z-sum in squash.
// ---------------------------------------------------------------------------
__global__ void __launch_bounds__(256) gemm_s_kernel(const float* __restrict__ x,
                                                     const float* __restrict__ wcs,
                                                     float* __restrict__ s_part) {
    const int ntile = blockIdx.x;          // 0..9
    const int mtile = blockIdx.y;          // 0..15
    const int z     = blockIdx.z;          // 0..KSPLIT-1
    const int tid  = threadIdx.x;
    const int wave = tid >> 5;
    const int lane = tid & 31;
    const int half = lane >> 4;
    const int ln   = lane & 15;

    const int m = mtile * 16 + ln;                 // A row for this lane
    const int chunk = z * 8 + wave;                // 0..15
    const int kb0 = chunk * KB_PER_WAVE;           // first K-block

    const float* aptr = x + (size_t)m * KDIM + kb0 * 4 + half * 2;
    const float* bptr = wcs + ((size_t)(ntile * KBLK + kb0) * 32 + lane) * 2;

    v8f acc = {};
#pragma unroll 4
    for (int it = 0; it < KB_PER_WAVE; ++it) {
        v2f a = *(const v2f*)aptr;                 // A[m,k0], A[m,k0+1]
        v2f b = *(const v2f*)bptr;                 // pre-swizzled B fragment
        acc = __builtin_amdgcn_wmma_f32_16x16x4_f32(
            false, a, false, b, (short)0, acc, false, false);
        aptr += 4;
        bptr += 64;
    }

    // deterministic cross-wave reduction of 8 partial 16x16 tiles
    __shared__ float red[8 * 256];
#pragma unroll
    for (int j = 0; j < 8; ++j) red[wave * 256 + lane * 8 + j] = acc[j];
    __syncthreads();
    float sum = 0.0f;
#pragma unroll
    for (int w = 0; w < 8; ++w) sum += red[w * 256 + tid];
    const int pl  = tid >> 3;              // source lane of this tile slot
    const int pj  = tid & 7;               // source vgpr index
    const int row = pj + 8 * (pl >> 4);    // D layout: row = j + 8*(lane>=16)
    const int col = pl & 15;
    s_part[(size_t)z * BATCH * NDIM +
           (size_t)(mtile * 16 + row) * NDIM + ntile * 16 + col] = sum;
}

// ---------------------------------------------------------------------------
// squash (elementwise in this reference): v = s^2*s / ((1+s^2)*sqrt(s^2))
// sums the KSPLIT partials in fixed order (deterministic).
// ---------------------------------------------------------------------------
__global__ void squash_kernel(const float* __restrict__ s_part,
                              float* __restrict__ v, int nelem) {
    int i = blockIdx.x * blockDim.x + threadIdx.x;
    if (i < nelem) {
        float sv = s_part[i];
#pragma unroll
        for (int z = 1; z < KSPLIT; ++z) sv += s_part[(size_t)z * nelem + i];
        float sq = sv * sv;
        v[i] = sq * sv / ((1.0f + sq) * sqrtf(sq));
    }
}

// ---------------------------------------------------------------------------
// G[r, i, (c,o)] = (1/B) * sum_b x[b,r,i] * v[b,(c,o)]
// One wave per (r, ntile): M=8 (zero-padded to 16), N=16, K=256.
// ---------------------------------------------------------------------------
__global__ void __launch_bounds__(32) gemm_g_kernel(const float* __restrict__ x,
                                                    const float* __restrict__ v,
                                                    float* __restrict__ g) {
    const int ntile = blockIdx.x;          // 0..9
    const int r     = blockIdx.y;          // 0..1151
    const int lane  = threadIdx.x & 31;
    const int half  = lane >> 4;
    const int ln    = lane & 15;
    const int n     = ntile * 16 + ln;

    const float* xcol = x + r * IN_CH + ln;   // valid only for ln < 8
    v8f acc = {};
#pragma unroll 4
    for (int k = half * 2; k < BATCH; k += 4) {
        v2f a, b;
        a.x = (ln < IN_CH) ? xcol[(size_t)k * KDIM] : 0.0f;       // A[i=ln, b=k]
        a.y = (ln < IN_CH) ? xcol[(size_t)(k + 1) * KDIM] : 0.0f;
        b.x = v[(size_t)k * NDIM + n];                            // B[b=k, n]
        b.y = v[(size_t)(k + 1) * NDIM + n];
        acc = __builtin_amdgcn_wmma_f32_16x16x4_f32(
            false, a, false, b, (short)0, acc, false, false);
    }
    // rows 0..7 of D live in lanes 0-15 (rows 8..15 are exact zeros; skip)
    if (lane < 16) {
#pragma unroll
        for (int j = 0; j < IN_CH; ++j)
            g[(size_t)(r * IN_CH + j) * NDIM + ntile * 16 + lane] =
                acc[j] * (1.0f / BATCH);
    }
}

// ---------------------------------------------------------------------------
// b_ij[r,c] += sum_{o,i} W[r,c,o,i] * G[r,i,(c,o)]
// ---------------------------------------------------------------------------
__global__ void bupdate_kernel(const float* __restrict__ W, const float* __restrict__ g,
                               float* __restrict__ b) {
    int t = blockIdx.x * blockDim.x + threadIdx.x;
    if (t >= RC) return;
    int r  = t / CAPS;
    int cc = t % CAPS;
    const float* wp = W + (size_t)t * OUT_CH * IN_CH;
    const float* gp = g + (size_t)r * IN_CH * NDIM + cc * OUT_CH;
    float sum = 0.0f;
    for (int o = 0; o < OUT_CH; ++o)
        for (int i = 0; i < IN_CH; ++i)
            sum += wp[o * IN_CH + i] * gp[(size_t)i * NDIM + o];
    b[t] += sum;
}

// ---------------------------------------------------------------------------
extern "C" void kernel_launch(void* const* d_in, const int* in_sizes, int n_in,
                              void* d_out, int out_size, void* d_ws, size_t ws_size,
                              hipStream_t stream) {
    (void)in_sizes; (void)n_in; (void)out_size; (void)ws_size;
    const float* x = (const float*)d_in[0];   // [256, 1152, 8]
    const float* W = (const float*)d_in[1];   // [1152, 10, 16, 8]
    float* out = (float*)d_out;               // [256, 10, 16, 1] flat

    float* ws     = (float*)d_ws;             // ~12.4 MB total, L2-resident
    float* b_ij   = ws;                       // 11520
    float* c_ij   = b_ij + RC;                // 11520
    float* s_part = c_ij + RC;                // KSPLIT * 40960
    float* v      = s_part + KSPLIT * BATCH * NDIM;   // 40960
    float* wcs    = v + BATCH * NDIM;         // 1474560 (B-fragment order)
    float* g      = wcs + (size_t)KDIM * NDIM;// 1474560

    zero_kernel<<<(RC + 255) / 256, 256, 0, stream>>>(b_ij, RC);

    for (int it = 0; it < NUM_ITER; ++it) {
        softmax_kernel<<<CAPS, 256, 0, stream>>>(b_ij, c_ij);
        scale_kernel<<<(KDIM * NDIM + 255) / 256, 256, 0, stream>>>(W, c_ij, wcs);
        gemm_s_kernel<<<dim3(CAPS, BATCH / 16, KSPLIT), 256, 0, stream>>>(
            x, wcs, s_part);
        float* vdst = (it == NUM_ITER - 1) ? out : v;
        squash_kernel<<<(BATCH * NDIM + 255) / 256, 256, 0, stream>>>(
            s_part, vdst, BATCH * NDIM);
        if (it < NUM_ITER - 1) {
            gemm_g_kernel<<<dim3(CAPS, ROUTES), 32, 0, stream>>>(x, v, g);
            bupdate_kernel<<<(RC + 255) / 256, 256, 0, stream>>>(W, g, b_ij);
        }
    }
}